// SimplifiedGCN_46866683134376
// MI455X (gfx1250) — compile-verified
//
#include <hip/hip_runtime.h>
#include <hip/hip_bf16.h>

typedef __attribute__((ext_vector_type(2))) float v2f;
typedef __attribute__((ext_vector_type(8))) float v8f;

#define DIM 64

// ---------------------------------------------------------------------------
// 1) in-degree of col (excluding self loop; +1 added in the transform below)
// ---------------------------------------------------------------------------
__global__ void gcn_deg_kernel(const int* __restrict__ col,
                               float* __restrict__ deg, int n_edges) {
    int e = blockIdx.x * blockDim.x + threadIdx.x;
    if (e < n_edges) {
        unsafeAtomicAdd(&deg[col[e]], 1.0f);
    }
}

// ---------------------------------------------------------------------------
// 2) deg -> deg_inv_sqrt in place: dis[i] = rsqrt(deg[i] + 1)   (self loop)
// ---------------------------------------------------------------------------
__global__ void gcn_dis_kernel(float* __restrict__ deg, int n_nodes) {
    int i = blockIdx.x * blockDim.x + threadIdx.x;
    if (i < n_nodes) {
        deg[i] = rsqrtf(deg[i] + 1.0f);
    }
}

// ---------------------------------------------------------------------------
// 3) propagation: hout[row] += hin[col] * dis[row]*dis[col]
//    virtual edges [n_edges, n_edges+n_nodes) are the self loops.
//    16 lanes per edge, float4 per lane (64 features).
// ---------------------------------------------------------------------------
__global__ void __launch_bounds__(256)
gcn_prop_kernel(const int* __restrict__ row, const int* __restrict__ col,
                const float* __restrict__ dis,
                const float* __restrict__ hin, float* __restrict__ hout,
                int n_edges, int n_nodes) {
    long long tid = (long long)blockIdx.x * blockDim.x + threadIdx.x;
    long long total = (long long)(n_edges + n_nodes) * 16;
    if (tid >= total) return;
    int e  = (int)(tid >> 4);
    int f4 = (int)(tid & 15);
    int r, c;
    if (e < n_edges) { r = row[e]; c = col[e]; }
    else             { r = c = e - n_edges; }
    float w = dis[r] * dis[c];
    const float4 v = ((const float4*)(hin + (long long)c * DIM))[f4];
    float* dst = hout + (long long)r * DIM + f4 * 4;
    unsafeAtomicAdd(dst + 0, v.x * w);
    unsafeAtomicAdd(dst + 1, v.y * w);
    unsafeAtomicAdd(dst + 2, v.z * w);
    unsafeAtomicAdd(dst + 3, v.w * w);
}

// ---------------------------------------------------------------------------
// 4) out = h @ W^T + b via V_WMMA_F32_16X16X4_F32 (full fp32 precision)
//    8 waves/block: waves 0-3 -> m-tile 0 (N tiles 0..3), waves 4-7 -> m-tile 1
//    Per wave: 16x16 output tile, K=64 -> 16 chained WMMAs.
//    A (16x4 f32): lane<16 holds K={k0,k0+1}, lane>=16 holds K={k0+2,k0+3}, M=lane&15
//    B (4x16 f32): N striped across lanes (N=lane&15), K rows {k0,k0+1}/{k0+2,k0+3}
//    D (16x16 f32): VGPR v, lane L -> M = v + 8*(L>>4), N = L&15
// ---------------------------------------------------------------------------
__global__ void __launch_bounds__(256)
gcn_gemm_wmma_kernel(const float* __restrict__ h, const float* __restrict__ W,
                     const float* __restrict__ bias, float* __restrict__ out,
                     int n_nodes) {
    int wave = threadIdx.x >> 5;   // 0..7
    int lane = threadIdx.x & 31;

    int mtile = blockIdx.x * 2 + (wave >> 2);
    int mBase = mtile * 16;
    int nBase = (wave & 3) * 16;
    if (mBase >= n_nodes) return;          // wave-uniform: EXEC stays all-1s

    int m    = mBase + (lane & 15);
    int n    = nBase + (lane & 15);
    int koff = (lane >> 4) * 2;            // 0 for lanes 0-15, 2 for lanes 16-31

    const float* arow = h + (long long)m * DIM + koff;   // h[m][koff + 4*kk ...]
    const float* brow = W + (long long)n * DIM + koff;   // W[n][k] == B[k][n]

    v8f acc = {};
#pragma unroll
    for (int kk = 0; kk < 16; ++kk) {
        v2f a = *(const v2f*)(arow + kk * 4);            // 8B-aligned
        v2f b = *(const v2f*)(brow + kk * 4);
        acc = __builtin_amdgcn_wmma_f32_16x16x4_f32(
            /*neg_a=*/false, a, /*neg_b=*/false, b,
            /*c_mod=*/(short)0, acc, /*reuse_a=*/false, /*reuse_b=*/false);
    }

    float bn   = bias[n];
    int   half = lane >> 4;
#pragma unroll
    for (int v = 0; v < 8; ++v) {
        int M = mBase + v + 8 * half;
        out[(long long)M * DIM + n] = acc[v] + bn;
    }
}

// ---------------------------------------------------------------------------
// host-side launch
// ---------------------------------------------------------------------------
extern "C" void kernel_launch(void* const* d_in, const int* in_sizes, int n_in,
                              void* d_out, int out_size, void* d_ws, size_t ws_size,
                              hipStream_t stream) {
    const float* x  = (const float*)d_in[0];
    const int*   ei = (const int*)d_in[1];
    const float* W  = (const float*)d_in[2];
    const float* b  = (const float*)d_in[3];

    int n_nodes = in_sizes[0] / DIM;
    int n_edges = in_sizes[1] / 2;
    const int* row = ei;            // ei[0] flattened first
    const int* col = ei + n_edges;  // ei[1]

    char* ws = (char*)d_ws;
    float* dis = (float*)ws;                                       // n_nodes f32
    size_t offA = (((size_t)n_nodes * 4) + 255) & ~(size_t)255;
    float* h1 = (float*)(ws + offA);                               // n_nodes*64 f32
    float* h2 = (float*)(ws + offA + (size_t)n_nodes * DIM * 4);   // n_nodes*64 f32

    hipMemsetAsync(dis, 0, (size_t)n_nodes * 4, stream);
    hipMemsetAsync(h1,  0, (size_t)n_nodes * DIM * 4, stream);
    hipMemsetAsync(h2,  0, (size_t)n_nodes * DIM * 4, stream);

    gcn_deg_kernel<<<(n_edges + 255) / 256, 256, 0, stream>>>(col, dis, n_edges);
    gcn_dis_kernel<<<(n_nodes + 255) / 256, 256, 0, stream>>>(dis, n_nodes);

    long long total = (long long)(n_edges + n_nodes) * 16;
    int pblocks = (int)((total + 255) / 256);
    gcn_prop_kernel<<<pblocks, 256, 0, stream>>>(row, col, dis, x,  h1, n_edges, n_nodes);
    gcn_prop_kernel<<<pblocks, 256, 0, stream>>>(row, col, dis, h1, h2, n_edges, n_nodes);

    int mtiles = (n_nodes + 15) / 16;          // 3125 for 50000
    int gblocks = (mtiles + 1) / 2;            // 2 m-tiles per block
    gcn_gemm_wmma_kernel<<<gblocks, 256, 0, stream>>>(h2, W, b, (float*)d_out, n_nodes);
}